// CBOW_1_68221260530031
// MI455X (gfx1250) — compile-verified
//
#include <hip/hip_runtime.h>

typedef __attribute__((ext_vector_type(2))) float v2f;
typedef __attribute__((ext_vector_type(8))) float v8f;

#define WAVES       4
#define DIM         128
#define CTX         10
#define KNEG        8
#define EX_PER_WAVE 16

// select c[v] for v in 0..7 without dynamic indexing (v_cndmask chain)
__device__ __forceinline__ float sel8(v8f c, int v) {
  float r = c[0];
  r = (v == 1) ? c[1] : r;
  r = (v == 2) ? c[2] : r;
  r = (v == 3) ? c[3] : r;
  r = (v == 4) ? c[4] : r;
  r = (v == 5) ? c[5] : r;
  r = (v == 6) ? c[6] : r;
  r = (v == 7) ? c[7] : r;
  return r;
}

__global__ __launch_bounds__(WAVES * 32)
void cbow_main(const int* __restrict__ contexts, const int* __restrict__ focus,
               const float* __restrict__ wmask, const float* __restrict__ labels,
               const float* __restrict__ ctx_emb, const float* __restrict__ neg_emb,
               float* __restrict__ partials) {
  // Per-wave private LDS tiles; waves never touch each other's region -> no barriers.
  __shared__ float s_src[WAVES][EX_PER_WAVE][DIM];   // 32 KB
  __shared__ float s_tgt[WAVES][EX_PER_WAVE][DIM];   // 32 KB

  const int tid  = threadIdx.x;
  const int wave = tid >> 5;
  const int lane = tid & 31;
  const int tile = blockIdx.x * WAVES + wave;
  const int base_b = tile * EX_PER_WAVE;

  // ---- Sum-pool context embeddings: coalesced 512B row reads (float4/lane) ----
  for (int b = 0; b < EX_PER_WAVE; ++b) {
    float4 acc = make_float4(0.f, 0.f, 0.f, 0.f);
    const int* crow = contexts + (long)(base_b + b) * CTX;  // uniform -> SMEM loads
#pragma unroll
    for (int c = 0; c < CTX; ++c) {
      const float4 e = *reinterpret_cast<const float4*>(
          ctx_emb + (long)crow[c] * DIM + lane * 4);
      acc.x += e.x; acc.y += e.y; acc.z += e.z; acc.w += e.w;
    }
    *reinterpret_cast<float4*>(&s_src[wave][b][lane * 4]) = acc;
  }

  // A-matrix lane mapping for V_WMMA_F32_16X16X4_F32:
  //   lanes 0-15:  vgpr0=K0, vgpr1=K1 ; lanes 16-31: vgpr0=K2, vgpr1=K3 (row M = lane&15)
  // B-matrix (4x16): row K striped across lanes within a VGPR (col N = lane&15).
  const int m    = lane & 15;
  const int half = lane >> 4;

  // Diagonal C[i][i] owners: i<8 -> lane=i, vgpr=i ; i>=8 -> lane=i+16, vgpr=i-8
  const bool owner = (lane < 8) || (lane >= 24);
  const int  b_own = (lane < 8) ? lane : (lane - 16);   // meaningful only if owner

  // ---- Preload per-pair BCE scalars once (owner lanes only): 2x float4 each ----
  float wmv[KNEG], lbv[KNEG];
#pragma unroll
  for (int k = 0; k < KNEG; ++k) { wmv[k] = 0.f; lbv[k] = 0.f; }
  if (owner) {
    const long roff = (long)(base_b + b_own) * KNEG;
    const float4 w0 = *reinterpret_cast<const float4*>(wmask + roff);
    const float4 w1 = *reinterpret_cast<const float4*>(wmask + roff + 4);
    const float4 l0 = *reinterpret_cast<const float4*>(labels + roff);
    const float4 l1 = *reinterpret_cast<const float4*>(labels + roff + 4);
    wmv[0] = w0.x; wmv[1] = w0.y; wmv[2] = w0.z; wmv[3] = w0.w;
    wmv[4] = w1.x; wmv[5] = w1.y; wmv[6] = w1.z; wmv[7] = w1.w;
    lbv[0] = l0.x; lbv[1] = l0.y; lbv[2] = l0.z; lbv[3] = l0.w;
    lbv[4] = l1.x; lbv[5] = l1.y; lbv[6] = l1.z; lbv[7] = l1.w;
  }

  float bce_sum = 0.f, w_sum = 0.f;

  // ---- Software-pipelined target staging: prefetch k+1 rows into registers
  //      while the WMMA chain for k consumes LDS. ----
  float4 r[EX_PER_WAVE];
#pragma unroll
  for (int b = 0; b < EX_PER_WAVE; ++b) {
    const int idx = focus[(long)(base_b + b) * KNEG + 0];  // uniform -> SMEM
    r[b] = *reinterpret_cast<const float4*>(neg_emb + (long)idx * DIM + lane * 4);
  }

#pragma unroll
  for (int k = 0; k < KNEG; ++k) {
    // commit prefetched rows for this k to LDS
#pragma unroll
    for (int b = 0; b < EX_PER_WAVE; ++b)
      *reinterpret_cast<float4*>(&s_tgt[wave][b][lane * 4]) = r[b];

    // issue gathers for k+1 now; latency hidden under the WMMA chain below
    if (k + 1 < KNEG) {
#pragma unroll
      for (int b = 0; b < EX_PER_WAVE; ++b) {
        const int idx = focus[(long)(base_b + b) * KNEG + (k + 1)];
        r[b] = *reinterpret_cast<const float4*>(neg_emb + (long)idx * DIM + lane * 4);
      }
    }

    // ---- pred[b,k] for 16 b's on the diagonal: 32 chained f32 WMMAs over D ----
    v8f cacc = {};
#pragma unroll
    for (int q = 0; q < DIM / 4; ++q) {
      const float2 a2 = *reinterpret_cast<const float2*>(
          &s_src[wave][m][4 * q + 2 * half]);           // ds_load_b64
      v2f a;  a[0] = a2.x; a[1] = a2.y;
      v2f bv;
      bv[0] = s_tgt[wave][m][4 * q + half];
      bv[1] = s_tgt[wave][m][4 * q + 2 + half];
      cacc = __builtin_amdgcn_wmma_f32_16x16x4_f32(
          false, a, false, bv, (short)0, cacc, false, false);
    }

    const int   v    = (lane < 8) ? lane : (lane - 24);
    const float pred = sel8(cacc, v & 7);

    if (owner) {
      // stable logaddexp(0, x) = max(x,0) + log1p(exp(-|x|))
      const float sp = fmaxf(pred, 0.f) + log1pf(__expf(-fabsf(pred)));
      bce_sum += wmv[k] * (sp - pred * lbv[k]);
      w_sum   += wmv[k];
    }
  }

  // per_row = sum_k bce / sum_k w ; reduce 16 owner lanes within the wave
  float pr = owner ? (bce_sum / w_sum) : 0.f;
#pragma unroll
  for (int off = 16; off >= 1; off >>= 1) pr += __shfl_down(pr, off, 32);
  if (lane == 0) partials[tile] = pr;
}

__global__ __launch_bounds__(256)
void cbow_reduce(const float* __restrict__ partials, float* __restrict__ out,
                 int n, float scale) {
  __shared__ float sm[256];
  float s = 0.f;
  for (int i = threadIdx.x; i < n; i += 256) s += partials[i];
  sm[threadIdx.x] = s;
  __syncthreads();
  for (int stride = 128; stride >= 1; stride >>= 1) {
    if ((int)threadIdx.x < stride) sm[threadIdx.x] += sm[threadIdx.x + stride];
    __syncthreads();
  }
  if (threadIdx.x == 0) out[0] = sm[0] * scale;
}

extern "C" void kernel_launch(void* const* d_in, const int* in_sizes, int n_in,
                              void* d_out, int out_size, void* d_ws, size_t ws_size,
                              hipStream_t stream) {
  const int*   contexts = (const int*)d_in[0];
  const int*   focus    = (const int*)d_in[1];
  const float* wmask    = (const float*)d_in[2];
  const float* labels   = (const float*)d_in[3];
  const float* ctx_emb  = (const float*)d_in[4];
  const float* neg_emb  = (const float*)d_in[5];

  const int B     = in_sizes[2] / KNEG;        // 16384
  const int tiles = B / EX_PER_WAVE;           // 1024 partials
  float* partials = (float*)d_ws;

  dim3 grid(B / (EX_PER_WAVE * WAVES));        // 256 blocks x 128 threads
  cbow_main<<<grid, WAVES * 32, 0, stream>>>(contexts, focus, wmask, labels,
                                             ctx_emb, neg_emb, partials);
  cbow_reduce<<<1, 256, 0, stream>>>(partials, (float*)d_out, tiles,
                                     1.0f / (float)B);
}